// DTRRouter_59184649339140
// MI455X (gfx1250) — compile-verified
//
#include <hip/hip_runtime.h>

// ---------------------------------------------------------------------------
// DTR router for MI455X (gfx1250, wave32).
//   Kernel 1: scores = hidden @ W + b   via V_WMMA_F32_16X16X4_F32 with
//             double-buffered GLOBAL_LOAD_ASYNC_TO_LDS_B128 staging
//             (ASYNCcnt path, DMA overlapped with WMMA, no intra-loop barriers)
//   Kernel 2: token_mask via exact rank count (reproduces argsort(argsort))
// ---------------------------------------------------------------------------

typedef __attribute__((ext_vector_type(2))) float v2f;
typedef __attribute__((ext_vector_type(4))) float v4f;
typedef __attribute__((ext_vector_type(8))) float v8f;

#define THREADS   256
#define WAVESZ    32
#define WAVES     (THREADS / WAVESZ)        // 8 waves / block
#define ROWS_PW   16                        // rows per wave (WMMA M)
#define ROWS_PB   (WAVES * ROWS_PW)         // 128 rows per block
#define KC        64                        // K-chunk staged in LDS
#define ROWSTRIDE 68                        // KC + 4 pad floats -> 4-bank skew per row
#define TILEF     (ROWS_PW * ROWSTRIDE)     // floats per wave tile
#define BUF_F     (WAVES * TILEF)           // floats per buffer (8704)
#define BUF_BYTES (BUF_F * 4)               // 34816 B

// ---------------------------------------------------------------------------
// Kernel 1: GEMV through the WMMA pipe, double-buffered async staging.
//   Wave w owns rows [blockRow + 16w, +16) and ONLY its own LDS slice, so no
//   per-chunk workgroup barriers are needed (Wlds is write-once + one barrier).
//   Steady state per chunk:
//     s_wait_asynccnt 0            (chunk n landed in buffer sel)
//     8x async b128 -> buffer !sel (chunk n+1 DMA in flight)
//     16x V_WMMA_F32_16X16X4_F32   (B = broadcast(W[k..k+3]) across N)
//   D[m][n] identical for every n => score of row m.
// ---------------------------------------------------------------------------
__global__ __launch_bounds__(THREADS)
void dtr_scores_wmma(const float* __restrict__ hidden,
                     const float* __restrict__ W,
                     const float* __restrict__ bias,
                     float* __restrict__ scores,
                     int C)
{
    __shared__ __align__(16) float Wlds[2048];     // full W, shared by block
    __shared__ __align__(16) float tile[2 * BUF_F];// double-buffered wave tiles

    const int tid     = threadIdx.x;
    const int lane    = tid & (WAVESZ - 1);
    const int w       = tid >> 5;
    const int half    = lane >> 4;          // 0: lanes 0-15, 1: lanes 16-31
    const int arow    = lane & 15;          // fragment row / staging column group
    const int rowBase = blockIdx.x * ROWS_PB + w * ROWS_PW;

    // Per-lane staging addresses, precomputed once.
    //   lds_slot[i]: LDS byte address (low 32 bits of generic ptr) in buffer 0
    //   grow_off[i]: byte offset of the 16B segment from `hidden` base (kc=0)
    unsigned lds_slot[ROWS_PW / 2];
    unsigned grow_off[ROWS_PW / 2];
    #pragma unroll
    for (int i = 0; i < ROWS_PW / 2; ++i) {
        const int r = 2 * i + half;
        lds_slot[i] = (unsigned)(size_t)&tile[w * TILEF + r * ROWSTRIDE + arow * 4];
        grow_off[i] = (unsigned)(((size_t)(rowBase + r) * (size_t)C
                                  + (size_t)(arow * 4)) * 4u);
    }

    // Kick off chunk 0 DMA into buffer 0 before anything else.
    #pragma unroll
    for (int i = 0; i < ROWS_PW / 2; ++i)
        asm volatile("global_load_async_to_lds_b128 %0, %1, %2"
                     :: "v"(lds_slot[i]), "v"(grow_off[i]), "s"(hidden)
                     : "memory");

    // Cache W in LDS (coalesced float4); one barrier covers its lifetime.
    for (int i = tid; i < (C >> 2); i += THREADS)
        *(v4f*)&Wlds[i * 4] = *(const v4f*)&W[i * 4];
    __syncthreads();

    // 4 independent accumulators to break the WMMA RAW chain.
    v8f acc0 = {}, acc1 = {}, acc2 = {}, acc3 = {};

    int sel = 0;
    for (int kc = 0; kc < C; kc += KC, sel ^= 1) {
        // Chunk kc has landed once ASYNCcnt drains (memory clobber keeps the
        // fragment ds_loads below this point).
        asm volatile("s_wait_asynccnt 0x0" ::: "memory");

        // Prefetch chunk kc+KC into the other buffer while we compute.
        if (kc + KC < C) {
            const unsigned lofs = (unsigned)((sel ^ 1) * BUF_BYTES);
            const unsigned gofs = (unsigned)((kc + KC) * 4);
            #pragma unroll
            for (int i = 0; i < ROWS_PW / 2; ++i) {
                unsigned laddr = lds_slot[i] + lofs;
                unsigned goff  = grow_off[i] + gofs;
                asm volatile("global_load_async_to_lds_b128 %0, %1, %2"
                             :: "v"(laddr), "v"(goff), "s"(hidden)
                             : "memory");
            }
        }

        const float* buf = &tile[sel * BUF_F + w * TILEF];

        // ---- 16 WMMAs per chunk (KC/4), 4-way interleaved ----
        #pragma unroll
        for (int kk = 0; kk < KC; kk += 16) {
            // A fragment (16x4 f32): lanes 0-15 hold K={kk,kk+1}, 16-31 K={kk+2,kk+3}
            // B fragment (4x16 f32): same K striping, value = W[k] broadcast over N
            const int ka = kk + 2 * half;
            v2f a0 = *(const v2f*)&buf[arow * ROWSTRIDE + ka + 0];
            v2f b0 = *(const v2f*)&Wlds[kc + ka + 0];
            acc0 = __builtin_amdgcn_wmma_f32_16x16x4_f32(false, a0, false, b0,
                                                         (short)0, acc0, false, false);
            v2f a1 = *(const v2f*)&buf[arow * ROWSTRIDE + ka + 4];
            v2f b1 = *(const v2f*)&Wlds[kc + ka + 4];
            acc1 = __builtin_amdgcn_wmma_f32_16x16x4_f32(false, a1, false, b1,
                                                         (short)0, acc1, false, false);
            v2f a2 = *(const v2f*)&buf[arow * ROWSTRIDE + ka + 8];
            v2f b2 = *(const v2f*)&Wlds[kc + ka + 8];
            acc2 = __builtin_amdgcn_wmma_f32_16x16x4_f32(false, a2, false, b2,
                                                         (short)0, acc2, false, false);
            v2f a3 = *(const v2f*)&buf[arow * ROWSTRIDE + ka + 12];
            v2f b3 = *(const v2f*)&Wlds[kc + ka + 12];
            acc3 = __builtin_amdgcn_wmma_f32_16x16x4_f32(false, a3, false, b3,
                                                         (short)0, acc3, false, false);
        }
        // Buffer `sel` is reused at chunk kc+2*KC, whose asyncs are issued
        // after this chunk's ds_loads have been s_wait_dscnt-drained by their
        // WMMA consumers (asm memory clobbers forbid reordering). Safe.
    }

    v8f acc = (acc0 + acc1) + (acc2 + acc3);
    const float bv = bias[0];

    // C/D layout: VGPR r = row (half ? 8+r : r) at lane column 0 / 16.
    if (lane == 0 || lane == 16) {
        const int mbase = rowBase + half * 8;
        #pragma unroll
        for (int r = 0; r < 8; ++r)
            scores[mbase + r] = acc[r] + bv;
    }
}

// ---------------------------------------------------------------------------
// Kernel 2: exact rank -> mask.
//   rank[t] = #{j : s[j] > s[t]}  +  #{j < t : s[j] == s[t]}
//   (identical to argsort(argsort(-s)) with stable ties)
//   mask[t] = rank[t] < max(1, int(clip(kr,0.1,1.0) * T))
// Grid: B * (T/256) blocks; each block keeps the whole score row in LDS.
// ---------------------------------------------------------------------------
__global__ __launch_bounds__(THREADS)
void dtr_topk_mask(const float* __restrict__ scores,
                   const float* __restrict__ keep_ratio,
                   float* __restrict__ mask,
                   int T, int segs)
{
    __shared__ __align__(16) float s[4096];

    const int b   = blockIdx.x / segs;
    const int seg = blockIdx.x - b * segs;
    const int tid = threadIdx.x;

    const float* srow = scores + (size_t)b * (size_t)T;
    for (int i = tid; i < (T >> 2); i += THREADS)
        *(v4f*)&s[i * 4] = *(const v4f*)&srow[i * 4];
    __syncthreads();

    const int   t  = seg * THREADS + tid;
    const float my = s[t];

    int cnt = 0;
    const v4f* s4 = (const v4f*)s;
    for (int j4 = 0; j4 < (T >> 2); ++j4) {
        const v4f v = s4[j4];              // LDS broadcast read, conflict-free
        const int j = j4 * 4;
        cnt += (v.x > my) || ((v.x == my) && (j + 0 < t));
        cnt += (v.y > my) || ((v.y == my) && (j + 1 < t));
        cnt += (v.z > my) || ((v.z == my) && (j + 2 < t));
        cnt += (v.w > my) || ((v.w == my) && (j + 3 < t));
    }

    float kr = keep_ratio[b];
    kr = fminf(1.0f, fmaxf(0.1f, kr));
    int kcnt = (int)(kr * (float)T);       // truncation, same as int() in ref
    if (kcnt < 1) kcnt = 1;

    mask[(size_t)b * (size_t)T + t] = (cnt < kcnt) ? 1.0f : 0.0f;
}

// ---------------------------------------------------------------------------
// Launch. Inputs (setup_inputs order): hidden [B,T,C] f32, keep_ratio [B] f32,
// W [C] f32, b [1] f32. Output: mask [B*T] then scores [B*T], both f32.
// ---------------------------------------------------------------------------
extern "C" void kernel_launch(void* const* d_in, const int* in_sizes, int n_in,
                              void* d_out, int out_size, void* d_ws, size_t ws_size,
                              hipStream_t stream) {
    const float* hidden     = (const float*)d_in[0];
    const float* keep_ratio = (const float*)d_in[1];
    const float* W          = (const float*)d_in[2];
    const float* bias       = (const float*)d_in[3];

    const int B = in_sizes[1];
    const int C = in_sizes[2];
    const int T = in_sizes[0] / (B * C);
    const int R = B * T;                       // 32768 rows

    float* mask   = (float*)d_out;
    float* scores = (float*)d_out + (size_t)R;

    dtr_scores_wmma<<<R / ROWS_PB, THREADS, 0, stream>>>(hidden, W, bias, scores, C);

    const int segs = T / THREADS;
    dtr_topk_mask<<<B * segs, THREADS, 0, stream>>>(scores, keep_ratio, mask, T, segs);
}